// SpikingResidualTemporalBlock_39642548142760
// MI455X (gfx1250) — compile-verified
//
#include <hip/hip_runtime.h>

// ---------------------------------------------------------------------------
// SpikingResidualTemporalBlock for MI455X (gfx1250, wave32).
//   GEMMs run on V_WMMA_F32_16X16X4_F32 (true fp32 matrix path — both GEMMs
//   feed LayerNorm -> hard spike threshold, so low-precision inputs would
//   flip spikes; fp32 WMMA keeps full precision on the matrix units).
//   Tiling: 32 rows x N per block; each wave owns a 32x32 tile (4 accums),
//   so per K-step the wave issues 4 b64 loads for 4 WMMAs (A fragments are
//   reused across 2 N-tiles, B fragments across 2 M-tiles).
//   LIF recurrences are sequential in T: separate coalesced scan kernels.
// ---------------------------------------------------------------------------

typedef __attribute__((ext_vector_type(2))) float v2f;
typedef __attribute__((ext_vector_type(8))) float v8f;

// Fused GEMM (+bias) + LayerNorm over N.
//   A:  [M, K] row-major          (activations)
//   Bw: [N, K] row-major          (weight, output-channel major: out = A * Bw^T)
//   out:[M, N] row-major, layernormed per row with gamma/beta.
// Block: 32 rows of M x all N columns. blockDim.x == N (N/32 waves,
// each wave computes four 16x16 WMMA tiles = 32 rows x 32 cols).
template<int N, int K>
__global__ __launch_bounds__(N) void gemm_bias_ln_kernel(
    const float* __restrict__ A,
    const float* __restrict__ Bw,
    const float* __restrict__ bias,
    const float* __restrict__ gamma,
    const float* __restrict__ beta,
    float* __restrict__ out)
{
    __shared__ float tile[32 * N];
    __shared__ float rsum[N];
    __shared__ float rsq[N];
    __shared__ float mrow[32];
    __shared__ float rrow[32];

    const int m0   = blockIdx.x * 32;
    const int lane = threadIdx.x & 31;
    const int wave = threadIdx.x >> 5;
    const int n0   = wave * 32;          // this wave's first output column
    const int nl   = lane & 15;
    const int koff = (lane >> 4) * 2;    // lanes 0-15 hold K0/K1, lanes 16-31 hold K2/K3

    // ISA f32 16x4 A layout: lane (l&15)=row M, VGPR j holds K = koff+j -> float2 load.
    const float* A0 = A  + (size_t)(m0 + nl) * K + koff;        // M-tile 0
    const float* A1 = A  + (size_t)(m0 + 16 + nl) * K + koff;   // M-tile 1
    // B is 4x16 (KxN) with B[k][n] = Bw[n][k]; same striping -> float2 from row n.
    const float* B0 = Bw + (size_t)(n0 + nl) * K + koff;        // N-tile 0
    const float* B1 = Bw + (size_t)(n0 + 16 + nl) * K + koff;   // N-tile 1

    v8f c00 = {};   // (M-tile0, N-tile0)
    v8f c01 = {};   // (M-tile0, N-tile1)
    v8f c10 = {};   // (M-tile1, N-tile0)
    v8f c11 = {};   // (M-tile1, N-tile1)
    #pragma unroll 4
    for (int k = 0; k < K; k += 4) {
        v2f a0 = *(const v2f*)(A0 + k);
        v2f a1 = *(const v2f*)(A1 + k);
        v2f b0 = *(const v2f*)(B0 + k);
        v2f b1 = *(const v2f*)(B1 + k);
        // (neg_a, A, neg_b, B, c_mod, C, reuse_a, reuse_b)
        c00 = __builtin_amdgcn_wmma_f32_16x16x4_f32(false, a0, false, b0,
                                                    (short)0, c00, false, false);
        c01 = __builtin_amdgcn_wmma_f32_16x16x4_f32(false, a0, false, b1,
                                                    (short)0, c01, false, false);
        c10 = __builtin_amdgcn_wmma_f32_16x16x4_f32(false, a1, false, b0,
                                                    (short)0, c10, false, false);
        c11 = __builtin_amdgcn_wmma_f32_16x16x4_f32(false, a1, false, b1,
                                                    (short)0, c11, false, false);
    }

    // C/D layout: VGPR v, lanes 0-15 -> (M=v, N=lane); lanes 16-31 -> (M=v+8, N=lane-16).
    const int   rbase = (lane >> 4) * 8;
    const float bia0  = bias[n0 + nl];
    const float bia1  = bias[n0 + 16 + nl];
    #pragma unroll
    for (int v = 0; v < 8; ++v) {
        tile[(rbase + v) * N + n0 + nl]           = c00[v] + bia0;
        tile[(rbase + v) * N + n0 + 16 + nl]      = c01[v] + bia1;
        tile[(16 + rbase + v) * N + n0 + nl]      = c10[v] + bia0;
        tile[(16 + rbase + v) * N + n0 + 16 + nl] = c11[v] + bia1;
    }
    __syncthreads();

    // LayerNorm over N per row; each row's columns are spread across waves, so
    // reduce through LDS. TPR threads cooperate on one row (32 rows, N threads).
    const int TPR = N / 32;
    const int row = threadIdx.x / TPR;
    const int sub = threadIdx.x % TPR;

    float s = 0.f, sq = 0.f;
    for (int n = sub; n < N; n += TPR) {
        float t = tile[row * N + n];
        s  += t;
        sq += t * t;
    }
    rsum[threadIdx.x] = s;
    rsq[threadIdx.x]  = sq;
    __syncthreads();

    if (sub == 0) {
        float ts = 0.f, tq = 0.f;
        for (int i = 0; i < TPR; ++i) {
            ts += rsum[row * TPR + i];
            tq += rsq[row * TPR + i];
        }
        float mean = ts * (1.0f / (float)N);
        float var  = tq * (1.0f / (float)N) - mean * mean;
        mrow[row] = mean;
        rrow[row] = rsqrtf(var + 1e-5f);
    }
    __syncthreads();

    const float mean = mrow[row];
    const float rstd = rrow[row];
    float* orow = out + (size_t)(m0 + row) * N;
    for (int n = sub; n < N; n += TPR) {
        float t = tile[row * N + n];
        orow[n] = (t - mean) * rstd * gamma[n] + beta[n];
    }
}

// LIF recurrence (tau=2, v_th=1, hard reset to 0), in place over [B, T, C].
// Block b handles batch b; thread = channel -> every timestep load/store is a
// fully coalesced C*4-byte burst. Each element is read once then overwritten.
template<int C>
__global__ __launch_bounds__(C) void lif_inplace_kernel(
    float* __restrict__ buf, int T)
{
    const size_t base = (size_t)blockIdx.x * (size_t)T * C + threadIdx.x;
    float v = 0.f;
    for (int t = 0; t < T; ++t) {
        const size_t idx = base + (size_t)t * C;
        float x  = buf[idx];
        float h  = 0.5f * (v + x);                 // v + (x - v)/tau, tau = 2
        float sp = (h >= 1.0f) ? 1.0f : 0.0f;      // Heaviside(h - v_th)
        v = h * (1.0f - sp);                       // hard reset
        buf[idx] = sp;
    }
}

// Second LIF fused with the residual add: out = x + spike(lif(oln)).
template<int C>
__global__ __launch_bounds__(C) void lif_residual_kernel(
    const float* __restrict__ oln,
    const float* __restrict__ xres,
    float* __restrict__ out, int T)
{
    const size_t base = (size_t)blockIdx.x * (size_t)T * C + threadIdx.x;
    float v = 0.f;
    for (int t = 0; t < T; ++t) {
        const size_t idx = base + (size_t)t * C;
        float x  = oln[idx];
        float h  = 0.5f * (v + x);
        float sp = (h >= 1.0f) ? 1.0f : 0.0f;
        v = h * (1.0f - sp);
        out[idx] = xres[idx] + sp;
    }
}

extern "C" void kernel_launch(void* const* d_in, const int* in_sizes, int n_in,
                              void* d_out, int out_size, void* d_ws, size_t ws_size,
                              hipStream_t stream)
{
    (void)in_sizes; (void)n_in; (void)out_size; (void)ws_size;

    const float* x   = (const float*)d_in[0];  // [B, T, D]
    const float* W1  = (const float*)d_in[1];  // [H, D]
    const float* b1  = (const float*)d_in[2];  // [H]
    const float* g1  = (const float*)d_in[3];
    const float* be1 = (const float*)d_in[4];
    const float* W2  = (const float*)d_in[5];  // [D, H]
    const float* b2  = (const float*)d_in[6];
    const float* g2  = (const float*)d_in[7];
    const float* be2 = (const float*)d_in[8];
    float* out = (float*)d_out;

    constexpr int B = 256, T = 256, D = 128, H = 256;
    constexpr int M = B * T;                       // 65536 rows

    float* h_buf = (float*)d_ws;                   // [M, H] fp32 (67 MB), reused in place for spikes
    float* o_buf = h_buf + (size_t)M * H;          // [M, D] fp32 (33.5 MB)

    // GEMM1 + bias + LN(H): h_buf = LN(x @ W1^T + b1)
    gemm_bias_ln_kernel<H, D><<<M / 32, H, 0, stream>>>(x, W1, b1, g1, be1, h_buf);

    // LIF over T, per (b, h-channel), spikes written in place.
    lif_inplace_kernel<H><<<B, H, 0, stream>>>(h_buf, T);

    // GEMM2 + bias + LN(D): o_buf = LN(spikes @ W2^T + b2)
    gemm_bias_ln_kernel<D, H><<<M / 32, D, 0, stream>>>(h_buf, W2, b2, g2, be2, o_buf);

    // LIF over T + residual: out = x + spikes2
    lif_residual_kernel<D><<<B, D, 0, stream>>>(o_buf, x, out, T);
}